// TaGSim_36867999269232
// MI455X (gfx1250) — compile-verified
//
#include <hip/hip_runtime.h>
#include <math.h>

typedef __attribute__((ext_vector_type(2))) float v2f;
typedef __attribute__((ext_vector_type(4))) float f4;
typedef __attribute__((ext_vector_type(8))) float v8f;

#define CHUNKS 32

struct PassArgs {
  const float* mat[4];
  const float* wt[4];
  int n[4];
  int rpc[4];
  int nmats;
  int maxn;
};

// ---------------------------------------------------------------------------
// Pass kernel: (optionally weighted) column sums of row-major n x n matrices.
//   out_partial[mi][chunk][col] = sum_{r in chunk} w[r] * M[r][col]
// Streaming 256MB matrices (> 192MB L2) -> non-temporal loads, float4 wide.
// ---------------------------------------------------------------------------
__global__ __launch_bounds__(256) void wcolsum_partial(PassArgs a,
                                                       float* __restrict__ partials) {
  const int mi = blockIdx.z;
  const float* __restrict__ M = a.mat[mi];
  const float* __restrict__ w = a.wt[mi];
  const int n = a.n[mi];
  const int col = blockIdx.x * 1024 + threadIdx.x * 4;
  const int chunk = blockIdx.y;
  if (col >= n) return;
  long r0 = (long)chunk * a.rpc[mi];
  long r1 = r0 + a.rpc[mi];
  if (r1 > n) r1 = n;
  float s0 = 0.f, s1 = 0.f, s2 = 0.f, s3 = 0.f;
  if ((n & 3) == 0) {
    const float* p = M + (size_t)r0 * (size_t)n + col;
    if (w) {
      for (long r = r0; r < r1; ++r, p += n) {
        f4 v = __builtin_nontemporal_load((const f4*)p);
        float ww = w[r];
        s0 = fmaf(ww, v.x, s0); s1 = fmaf(ww, v.y, s1);
        s2 = fmaf(ww, v.z, s2); s3 = fmaf(ww, v.w, s3);
      }
    } else {
      for (long r = r0; r < r1; ++r, p += n) {
        f4 v = __builtin_nontemporal_load((const f4*)p);
        s0 += v.x; s1 += v.y; s2 += v.z; s3 += v.w;
      }
    }
  } else {  // generic fallback (not hit for n=8000)
    for (long r = r0; r < r1; ++r) {
      const float* rp = M + (size_t)r * (size_t)n;
      float ww = w ? w[r] : 1.0f;
      s0 = fmaf(ww, rp[col], s0);
      if (col + 1 < n) s1 = fmaf(ww, rp[col + 1], s1);
      if (col + 2 < n) s2 = fmaf(ww, rp[col + 2], s2);
      if (col + 3 < n) s3 = fmaf(ww, rp[col + 3], s3);
    }
  }
  float* P = partials + ((size_t)mi * CHUNKS + chunk) * (size_t)a.maxn + col;
  P[0] = s0;
  if (col + 1 < n) P[1] = s1;
  if (col + 2 < n) P[2] = s2;
  if (col + 3 < n) P[3] = s3;
}

// Deterministic fixed-order reduction of the row-chunk partials.
__global__ __launch_bounds__(256) void reduce_partials(const float* __restrict__ partials,
                                                       float* __restrict__ out,
                                                       PassArgs a) {
  int idx = blockIdx.x * 256 + threadIdx.x;
  int mi = idx / a.maxn;
  int col = idx - mi * a.maxn;
  if (mi >= a.nmats || col >= a.n[mi]) return;
  float s = 0.f;
  for (int ch = 0; ch < CHUNKS; ++ch)
    s += partials[((size_t)mi * CHUNKS + ch) * (size_t)a.maxn + col];
  out[(size_t)mi * a.maxn + col] = s;
}

// ---------------------------------------------------------------------------
// Feature pooling: builds the six pooled vectors
//   vec0=g1_01  vec1=g2_01  vec2=g1_12  vec3=g2_12  vec4=e1_01  vec5=e2_01
// stored at vecs + 32*i.  fs = F.sum(0), h1s = c.F, h2s = d.F
// ---------------------------------------------------------------------------
__global__ __launch_bounds__(256) void pool_features(
    const float* __restrict__ F1, const float* __restrict__ F2,
    const float* __restrict__ EF1, const float* __restrict__ EF2,
    const float* __restrict__ cvec, const float* __restrict__ dvec,
    int N, int E, int maxn, float* __restrict__ vecs) {
  const int bid = blockIdx.x;
  const int t = threadIdx.x;
  const float* F; const float* c; const float* d = nullptr;
  int nf, n; float* v01; float* v12 = nullptr;
  if (bid == 0)      { F = F1;  c = cvec;                    d = dvec;        nf = 16; n = N; v01 = vecs + 0;   v12 = vecs + 64; }
  else if (bid == 1) { F = F2;  c = cvec + maxn;             d = dvec + maxn; nf = 16; n = N; v01 = vecs + 32;  v12 = vecs + 96; }
  else if (bid == 2) { F = EF1; c = cvec + 2 * (size_t)maxn;                  nf = 4;  n = E; v01 = vecs + 128; }
  else               { F = EF2; c = cvec + 3 * (size_t)maxn;                  nf = 4;  n = E; v01 = vecs + 160; }
  const int f = t % nf;
  const int g = t / nf;
  const int ng = 256 / nf;
  float as = 0.f, ac = 0.f, ad = 0.f;
  for (int m = g; m < n; m += ng) {
    float x = F[(size_t)m * nf + f];
    as += x;
    ac = fmaf(c[m], x, ac);
    if (d) ad = fmaf(d[m], x, ad);
  }
  __shared__ float red[3][256];
  red[0][t] = as; red[1][t] = ac; red[2][t] = ad;
  __syncthreads();
  if (t < nf) {
    float S = 0.f, C = 0.f, D = 0.f;
    for (int gg = 0; gg < ng; ++gg) {
      S += red[0][gg * nf + t];
      C += red[1][gg * nf + t];
      D += red[2][gg * nf + t];
    }
    v01[t] = S; v01[nf + t] = C;
    if (v12) { v12[t] = C; v12[nf + t] = D; }
  }
}

// ---------------------------------------------------------------------------
// Branch head: bilinear tensor via V_WMMA_F32_16X16X4_F32 + tiny MLP.
// scoring[k] = sum_p Ofl[p] * Wflat[p*16+k],  Ofl = outer(e1,e2) flattened.
// One wave32 per branch; EXEC is all-ones around every WMMA.
// A layout (16x4 f32): v0 = K{0|2}, v1 = K{1|3}, lane halves = K pairs.
// B layout (4x16 f32): v0 = rows K{0|2} striped on lanes, v1 = rows K{1|3}.
// Only A row M=0 is nonzero -> D row 0 (c[0], lanes 0..15) = scoring[k].
// ---------------------------------------------------------------------------
struct BranchParams {
  const float *W, *Wb, *b, *w1, *b1, *w2, *b2, *w3, *b3, *w4, *b4;
};
struct AllBranches { BranchParams p[4]; };

__global__ __launch_bounds__(32) void branch_kernel(AllBranches ab,
                                                    const float* __restrict__ vecs,
                                                    float* __restrict__ out) {
  const int b = blockIdx.x;
  const BranchParams P = ab.p[b];
  const int dim = (b == 3) ? 8 : 32;
  const float* eA = vecs + ((b <= 1) ? 0 : (b == 2) ? 64 : 128);
  const float* eB = eA + 32;
  const int lane = threadIdx.x;

  __shared__ float cat[64];      // [e1 ; e2]
  __shared__ float Ofl[1024];    // outer(e1,e2) flattened, dim*dim entries
  __shared__ float sbuf[16], h1s[16], h2s[8], h3s[4];

  if (lane < dim) { cat[lane] = eA[lane]; cat[dim + lane] = eB[lane]; }
  __syncthreads();
  const int dd = dim * dim;
  for (int p = lane; p < dd; p += 32)
    Ofl[p] = cat[p / dim] * cat[dim + (p % dim)];
  __syncthreads();

  v8f acc = {0.f, 0.f, 0.f, 0.f, 0.f, 0.f, 0.f, 0.f};
  const int half = lane >> 4;
  const int l15 = lane & 15;
  const int nch = dd >> 2;
  const float* __restrict__ Wf = P.W;
  for (int ch = 0; ch < nch; ++ch) {
    const int p0 = ch * 4 + 2 * half;
    float a0 = (l15 == 0) ? Ofl[p0] : 0.0f;      // A[0,K]
    float a1 = (l15 == 0) ? Ofl[p0 + 1] : 0.0f;
    v2f A = {a0, a1};
    v2f B = {Wf[p0 * 16 + l15], Wf[(p0 + 1) * 16 + l15]};
    acc = __builtin_amdgcn_wmma_f32_16x16x4_f32(false, A, false, B,
                                                (short)0, acc, false, false);
  }
  float scoring = acc[0];  // D[M=0, N=lane] for lanes 0..15

  if (lane < 16) {
    float s = P.b[lane];
    const int tw = 2 * dim;
    for (int t = 0; t < tw; ++t) s = fmaf(P.Wb[lane * tw + t], cat[t], s);
    s += scoring;
    sbuf[lane] = s > 0.f ? s : 0.f;
  }
  __syncthreads();
  if (lane < 16) {
    float a = P.b1[lane];
    for (int t = 0; t < 16; ++t) a = fmaf(sbuf[t], P.w1[t * 16 + lane], a);
    h1s[lane] = a > 0.f ? a : 0.f;
  }
  __syncthreads();
  if (lane < 8) {
    float a = P.b2[lane];
    for (int t = 0; t < 16; ++t) a = fmaf(h1s[t], P.w2[t * 8 + lane], a);
    h2s[lane] = a > 0.f ? a : 0.f;
  }
  __syncthreads();
  if (lane < 4) {
    float a = P.b3[lane];
    for (int t = 0; t < 8; ++t) a = fmaf(h2s[t], P.w3[t * 4 + lane], a);
    h3s[lane] = a > 0.f ? a : 0.f;
  }
  __syncthreads();
  if (lane == 0) {
    float a = P.b4[0];
    for (int t = 0; t < 4; ++t) a = fmaf(h3s[t], P.w4[t], a);
    out[b] = 1.0f / (1.0f + expf(-a));
  }
}

// ---------------------------------------------------------------------------
static void fill_concat(BranchParams* bp, const float* base, int dim) {
  size_t o = 0;
  bp->W  = base + o; o += (size_t)dim * dim * 16;
  bp->Wb = base + o; o += (size_t)16 * 2 * dim;
  bp->b  = base + o; o += 16;
  bp->w1 = base + o; o += 16 * 16;
  bp->b1 = base + o; o += 16;
  bp->w2 = base + o; o += 16 * 8;
  bp->b2 = base + o; o += 8;
  bp->w3 = base + o; o += 8 * 4;
  bp->b3 = base + o; o += 4;
  bp->w4 = base + o; o += 4;
  bp->b4 = base + o; o += 1;
}

extern "C" void kernel_launch(void* const* d_in, const int* in_sizes, int n_in,
                              void* d_out, int out_size, void* d_ws, size_t ws_size,
                              hipStream_t stream) {
  (void)out_size; (void)ws_size;
  const float* adj1  = (const float*)d_in[0];
  const float* adj2  = (const float*)d_in[1];
  const float* eadj1 = (const float*)d_in[2];
  const float* eadj2 = (const float*)d_in[3];
  const float* F1  = (const float*)d_in[4];
  const float* F2  = (const float*)d_in[5];
  const float* EF1 = (const float*)d_in[6];
  const float* EF2 = (const float*)d_in[7];
  const int N = in_sizes[4] / 16;   // node count
  const int E = in_sizes[6] / 4;    // edge count
  const int maxn = (N > E) ? N : E;

  float* ws = (float*)d_ws;
  size_t off = 0;
  float* partials = ws + off; off += (size_t)4 * CHUNKS * maxn;
  float* cvec     = ws + off; off += (size_t)4 * maxn;   // c1,c2,ce1,ce2
  float* dvec     = ws + off; off += (size_t)2 * maxn;   // d1,d2
  float* vecs     = ws + off; off += 256;                // 6 pooled vectors

  // Pass 1: colsums of all 4 adjacency matrices.
  PassArgs a1{};
  a1.mat[0] = adj1; a1.mat[1] = adj2; a1.mat[2] = eadj1; a1.mat[3] = eadj2;
  a1.wt[0] = a1.wt[1] = a1.wt[2] = a1.wt[3] = nullptr;
  a1.n[0] = a1.n[1] = N; a1.n[2] = a1.n[3] = E;
  for (int i = 0; i < 4; ++i) a1.rpc[i] = (a1.n[i] + CHUNKS - 1) / CHUNKS;
  a1.nmats = 4; a1.maxn = maxn;
  dim3 g1((maxn + 1023) / 1024, CHUNKS, 4);
  wcolsum_partial<<<g1, dim3(256), 0, stream>>>(a1, partials);
  int tot1 = 4 * maxn;
  reduce_partials<<<dim3((tot1 + 255) / 256), dim3(256), 0, stream>>>(partials, cvec, a1);

  // Pass 2: d = c^T @ adj for the two node graphs (needs c fully reduced).
  PassArgs a2{};
  a2.mat[0] = adj1; a2.mat[1] = adj2;
  a2.wt[0] = cvec;  a2.wt[1] = cvec + maxn;
  a2.n[0] = a2.n[1] = N;
  a2.rpc[0] = a2.rpc[1] = (N + CHUNKS - 1) / CHUNKS;
  a2.nmats = 2; a2.maxn = maxn;
  dim3 g2((N + 1023) / 1024, CHUNKS, 2);
  wcolsum_partial<<<g2, dim3(256), 0, stream>>>(a2, partials);
  int tot2 = 2 * maxn;
  reduce_partials<<<dim3((tot2 + 255) / 256), dim3(256), 0, stream>>>(partials, dvec, a2);

  // Pooled vectors.
  pool_features<<<dim3(4), dim3(256), 0, stream>>>(F1, F2, EF1, EF2, cvec, dvec,
                                                   N, E, maxn, vecs);

  // Branch parameters: concat-per-branch (n_in==12) or one pointer per leaf.
  AllBranches ab{};
  if (n_in == 12) {
    const int dims[4] = {32, 32, 32, 8};
    for (int i = 0; i < 4; ++i)
      fill_concat(&ab.p[i], (const float*)d_in[8 + i], dims[i]);
  } else {
    int per = (n_in - 8) / 4;  // expected 11 leaves per branch, insertion order
    for (int i = 0; i < 4; ++i) {
      int base = 8 + i * per;
      ab.p[i].W  = (const float*)d_in[base + 0];
      ab.p[i].Wb = (const float*)d_in[base + 1];
      ab.p[i].b  = (const float*)d_in[base + 2];
      ab.p[i].w1 = (const float*)d_in[base + 3];
      ab.p[i].b1 = (const float*)d_in[base + 4];
      ab.p[i].w2 = (const float*)d_in[base + 5];
      ab.p[i].b2 = (const float*)d_in[base + 6];
      ab.p[i].w3 = (const float*)d_in[base + 7];
      ab.p[i].b3 = (const float*)d_in[base + 8];
      ab.p[i].w4 = (const float*)d_in[base + 9];
      ab.p[i].b4 = (const float*)d_in[base + 10];
    }
  }
  branch_kernel<<<dim3(4), dim3(32), 0, stream>>>(ab, vecs, (float*)d_out);
}